// DeepProbLogAdditionReasoner_52192442581033
// MI455X (gfx1250) — compile-verified
//
#include <hip/hip_runtime.h>

// D = A(16x4 f32) * B(4x16 f32) + C(16x16 f32), wave32 WMMA.
typedef __attribute__((ext_vector_type(2))) float v2f;
typedef __attribute__((ext_vector_type(8))) float v8f;

#define EPS_F 1e-9f

// Compile-time (after full unroll) digit helpers for flattened m = 10*i + j.
constexpr int DI(int m)   { return m / 10; }
constexpr int DJ(int m)   { return m % 10; }
constexpr int DSUM(int m) { return m / 10 + m % 10; }

__global__ __launch_bounds__(256)
void dpl_add_wmma_kernel(const float* __restrict__ p1,
                         const float* __restrict__ p2,
                         float* __restrict__ out, int B)
{
    // Per-wave scratch: 16 rows x (19 bins + inv) with stride 20 (bank-conflict free).
    __shared__ float lds[8][16][20];

    const int tid  = threadIdx.x;
    const int lane = tid & 31;
    const int w    = tid >> 5;            // wave within block
    const int col  = lane & 15;           // N / row-in-tile index
    const bool lo  = lane < 16;           // lower half-wave?

    const int  tile   = blockIdx.x * 8 + w;
    const int  b0     = tile * 16;        // wave-uniform
    const bool active = b0 < B;           // wave-uniform -> EXEC stays all-1s inside

    if (active) {
        int brow = b0 + col;
        if (brow >= B) brow = B - 1;      // clamp for (never-hit) partial tile
        const float* __restrict__ r1 = p1 + (size_t)brow * 10;
        const float* __restrict__ r2 = p2 + (size_t)brow * 10;

        float p1v[10], p2v[10];
        #pragma unroll
        for (int k = 0; k < 10; ++k) { p1v[k] = r1[k]; p2v[k] = r2[k]; }

        // Bins 16..18 (outside the 16-wide WMMA N-tile): only 6 terms, do directly.
        float t16 = p1v[7]*p2v[9] + p1v[8]*p2v[8] + p1v[9]*p2v[7];
        float t17 = p1v[8]*p2v[9] + p1v[9]*p2v[8];
        float t18 = p1v[9]*p2v[9];

        // WMMA over K = 100 (25 chunks of 4): D[16 rows, 16 bins] += T_chunk * M_chunk.
        // A 16x4 layout: lanes 0-15 hold (K=0 in v0, K=1 in v1); lanes 16-31 hold (K=2, K=3).
        // B 4x16 layout (mirror): v0 = row K0 (low lanes) / K2 (high lanes), v1 = K1 / K3.
        v8f acc = {0.f, 0.f, 0.f, 0.f, 0.f, 0.f, 0.f, 0.f};
        #pragma unroll
        for (int kk = 0; kk < 25; ++kk) {
            const int m0 = kk * 4;
            v2f a, b;
            // T[brow, m] = p1[m/10] * p2[m%10]; indices constant-fold after unroll.
            a.x = lo ? p1v[DI(m0+0)] * p2v[DJ(m0+0)]
                     : p1v[DI(m0+2)] * p2v[DJ(m0+2)];
            a.y = lo ? p1v[DI(m0+1)] * p2v[DJ(m0+1)]
                     : p1v[DI(m0+3)] * p2v[DJ(m0+3)];
            // M[m, n] = (digit_sum(m) == n); bins >15 handled by the tail path.
            int s0 = lo ? DSUM(m0+0) : DSUM(m0+2);
            int s1 = lo ? DSUM(m0+1) : DSUM(m0+3);
            b.x = (col == s0) ? 1.0f : 0.0f;
            b.y = (col == s1) ? 1.0f : 0.0f;
            // 8 args: (neg_a, A, neg_b, B, c_mod, C, reuse_a, reuse_b)
            acc = __builtin_amdgcn_wmma_f32_16x16x4_f32(
                      false, a, false, b, (short)0, acc, false, false);
        }

        // D layout: VGPR v -> (row = v + (lo?0:8), bin = col). Scatter to LDS.
        const int rbase = lo ? 0 : 8;
        #pragma unroll
        for (int v = 0; v < 8; ++v)
            lds[w][rbase + v][col] = acc[v];
        if (lo) {
            lds[w][lane][16] = t16;
            lds[w][lane][17] = t17;
            lds[w][lane][18] = t18;
        }
    }
    __syncthreads();

    // Per-row normalization factor.
    if (active && lo) {
        float s = 0.f;
        #pragma unroll
        for (int k = 0; k < 19; ++k) s += lds[w][lane][k];
        lds[w][lane][19] = 1.0f / (s + EPS_F);
    }
    __syncthreads();

    // Cooperative, fully-coalesced store of the wave's contiguous 16*19 outputs.
    if (active) {
        float* __restrict__ outw = out + (size_t)b0 * 19;
        int total = (B - b0) * 19;
        if (total > 304) total = 304;
        #pragma unroll
        for (int it = 0; it < 10; ++it) {
            int idx = it * 32 + lane;
            if (idx < total) {
                unsigned row = (unsigned)idx / 19u;   // magic-mul division
                unsigned c   = (unsigned)idx - row * 19u;
                outw[idx] = lds[w][row][c] * lds[w][row][19];
            }
        }
    }
}

extern "C" void kernel_launch(void* const* d_in, const int* in_sizes, int n_in,
                              void* d_out, int out_size, void* d_ws, size_t ws_size,
                              hipStream_t stream) {
    const float* p1 = (const float*)d_in[0];
    const float* p2 = (const float*)d_in[1];
    float* out = (float*)d_out;
    const int B = in_sizes[0] / 10;           // [B,10] fp32

    const int tiles  = (B + 15) / 16;         // 16 batch rows per wave
    const int blocks = (tiles + 7) / 8;       // 8 waves (256 threads) per block
    dpl_add_wmma_kernel<<<blocks, 256, 0, stream>>>(p1, p2, out, B);
}